// GCN_7842610282622
// MI455X (gfx1250) — compile-verified
//
#include <hip/hip_runtime.h>
#include <hip/hip_bf16.h>

// Problem constants (from reference): B, N, E, F, H = 8, 20000, 640000, 8, 128
#define BATCH 8
#define NODES 20000
#define EDGES 640000
#define FEAT  8
#define HID   128
#define CAP   512   // match-list capacity per graph; E[K]=E/N=32, Poisson -> P(K>512) ~ 0

typedef float v2f __attribute__((ext_vector_type(2)));
typedef float v8f __attribute__((ext_vector_type(8)));

// ---------------- Kernel A: zero degree counters + match counters -------------
__global__ void gcn_zero(int* __restrict__ deg, int* __restrict__ cnt) {
    int t = blockIdx.x * blockDim.x + threadIdx.x;
    if (t < BATCH * NODES) deg[t] = 0;
    if (t < BATCH) cnt[t] = 0;
}

// ---------------- Kernel B: stream all edges once -----------------------------
// deg[b][v] = #edges with dst==v ; collect edges with dst == node_idx[b]
__global__ void gcn_scan(const int* __restrict__ edge_index,
                         const int* __restrict__ node_idx,
                         int* __restrict__ deg, int* __restrict__ cnt,
                         int* __restrict__ me, int* __restrict__ msrc) {
    int e = blockIdx.x * blockDim.x + threadIdx.x;
    int b = blockIdx.y;
    if (e >= EDGES) return;
    const int* ei = edge_index + (size_t)b * 2 * EDGES;
    int dst = ei[EDGES + e];
    atomicAdd(&deg[b * NODES + dst], 1);
    if (dst == node_idx[b]) {
        int src = ei[e];
        int pos = atomicAdd(&cnt[b], 1);
        if (pos < CAP) { me[b * CAP + pos] = e; msrc[b * CAP + pos] = src; }
    }
}

// ---------------- WMMA f32 MLP layer: out(1x128) = relu(in(1x128) @ W + bias) -
// One wave per 16-column tile (8 waves cover H=128); K-loop of V_WMMA_F32_16X16X4_F32.
__device__ __forceinline__ void mlp_layer_wmma(const float* __restrict__ W,
                                               const float* __restrict__ bias,
                                               const float* __restrict__ s_in,
                                               float* __restrict__ s_out, int tid) {
    const int wave = tid >> 5;
    const int lane = tid & 31;
    const int j0 = wave * 16;          // output column tile
    v8f c = {};
    for (int k0 = 0; k0 < HID; k0 += 4) {
        // A (16x4 f32): only row M=0 non-zero. lane0 -> (K=k0,k0+1); lane16 -> (K=k0+2,k0+3)
        v2f a;
        a[0] = (lane == 0) ? s_in[k0]     : ((lane == 16) ? s_in[k0 + 2] : 0.0f);
        a[1] = (lane == 0) ? s_in[k0 + 1] : ((lane == 16) ? s_in[k0 + 3] : 0.0f);
        // B (4x16 f32): VGPR r holds rows K=r (lanes 0-15) and K=r+2 (lanes 16-31)
        const int krow = (lane < 16) ? k0 : (k0 + 2);
        const int n    = j0 + (lane & 15);
        v2f bf;
        bf[0] = W[(size_t)krow * HID + n];
        bf[1] = W[(size_t)(krow + 1) * HID + n];
        c = __builtin_amdgcn_wmma_f32_16x16x4_f32(false, a, false, bf,
                                                  (short)0, c, false, false);
    }
    // D (16x16 f32): row M=0 lives in VGPR0, lanes 0-15 (N = j0 + lane)
    if (lane < 16) {
        float v = c[0] + bias[j0 + lane];
        s_out[j0 + lane] = fmaxf(v, 0.0f);
    }
}

// ---------------- Kernel C: per-graph epilogue (1 block / graph, 256 thr) -----
__global__ void gcn_finalize(const float* __restrict__ x,
                             const int* __restrict__ node_idx,
                             const float* __restrict__ W1, const float* __restrict__ b1,
                             const float* __restrict__ W2, const float* __restrict__ b2,
                             const float* __restrict__ W3, const float* __restrict__ b3,
                             const float* __restrict__ Wl, const float* __restrict__ bl,
                             const int* __restrict__ deg, const int* __restrict__ cnt,
                             const int* __restrict__ me,  const int* __restrict__ msrc,
                             float* __restrict__ out) {
    const int b = blockIdx.x;
    const int tid = threadIdx.x;

    __shared__ int   s_e[CAP];
    __shared__ int   s_src[CAP];      // sorted by edge id
    __shared__ float s_coef[CAP];     // sorted by edge id
    __shared__ float s_xrow[CAP][FEAT];
    __shared__ float s_h0[HID];
    __shared__ float s_h1[HID];
    __shared__ float s_h2[HID];
    __shared__ int   s_K;
    __shared__ float s_dinvn;

    const int nid = node_idx[b];
    if (tid == 0) {
        int K = cnt[b]; if (K > CAP) K = CAP;
        s_K = K;
        s_dinvn = rsqrtf((float)(deg[b * NODES + nid] + 1));
    }
    __syncthreads();
    const int K = s_K;
    const float dinvn = s_dinvn;

    for (int i = tid; i < K; i += 256) s_e[i] = me[b * CAP + i];
    __syncthreads();

    // Deterministic order: rank by unique edge id, scatter src/coef to sorted slots
    for (int i = tid; i < K; i += 256) {
        int ei = s_e[i];
        int r = 0;
        for (int j = 0; j < K; ++j) r += (s_e[j] < ei) ? 1 : 0;
        int src = msrc[b * CAP + i];
        s_src[r]  = src;
        s_coef[r] = rsqrtf((float)(deg[b * NODES + src] + 1)) * dinvn;
    }
    __syncthreads();

    // Stage matched x rows (F=8 each) into LDS
    for (int t = tid; t < K * FEAT; t += 256) {
        int i = t >> 3, f = t & 7;
        s_xrow[i][f] = x[((size_t)b * NODES + s_src[i]) * FEAT + f];
    }
    __syncthreads();

    // Layer 1 (GCNConv at node nid): channel j per thread, sequential (deterministic) sum
    if (tid < HID) {
        const int j = tid;
        float w[FEAT];
#pragma unroll
        for (int f = 0; f < FEAT; ++f) w[f] = W1[f * HID + j];
        float self = 0.0f;
#pragma unroll
        for (int f = 0; f < FEAT; ++f)
            self += x[((size_t)b * NODES + nid) * FEAT + f] * w[f];
        float acc = self * dinvn * dinvn;
        for (int i = 0; i < K; ++i) {
            float d = 0.0f;
#pragma unroll
            for (int f = 0; f < FEAT; ++f) d += s_xrow[i][f] * w[f];
            acc += s_coef[i] * d;
        }
        acc += b1[j];
        s_h0[j] = fmaxf(acc, 0.0f);
    }
    __syncthreads();

    // Layers 2 & 3 via V_WMMA_F32_16X16X4_F32 (all 8 waves, EXEC all-ones)
    mlp_layer_wmma(W2, b2, s_h0, s_h1, tid);
    __syncthreads();
    mlp_layer_wmma(W3, b3, s_h1, s_h2, tid);
    __syncthreads();

    // Head: logits = h2 @ Wl + bl  (2 outputs)
    if (tid < 2) {
        float acc = bl[tid];
        for (int k = 0; k < HID; ++k) acc += s_h2[k] * Wl[k * 2 + tid];
        out[b * 2 + tid] = acc;
    }
}

// ---------------------------------------------------------------------------
extern "C" void kernel_launch(void* const* d_in, const int* in_sizes, int n_in,
                              void* d_out, int out_size, void* d_ws, size_t ws_size,
                              hipStream_t stream) {
    const float* x    = (const float*)d_in[0];
    const int* eidx   = (const int*)d_in[1];
    const int* nidx   = (const int*)d_in[2];
    const float* W1   = (const float*)d_in[3];
    const float* b1   = (const float*)d_in[4];
    const float* W2   = (const float*)d_in[5];
    const float* b2   = (const float*)d_in[6];
    const float* W3   = (const float*)d_in[7];
    const float* b3   = (const float*)d_in[8];
    const float* Wl   = (const float*)d_in[9];
    const float* bl   = (const float*)d_in[10];
    float* out        = (float*)d_out;

    // Workspace layout (ints): deg[B*N] | cnt[B..pad 64] | me[B*CAP] | msrc[B*CAP]
    int* deg  = (int*)d_ws;
    int* cnt  = deg + BATCH * NODES;
    int* me   = cnt + 64;
    int* msrc = me + BATCH * CAP;   // total ~673 KB

    {
        int total = BATCH * NODES;
        gcn_zero<<<(total + 255) / 256, 256, 0, stream>>>(deg, cnt);
    }
    {
        dim3 grid(EDGES / 256, BATCH);
        gcn_scan<<<grid, 256, 0, stream>>>(eidx, nidx, deg, cnt, me, msrc);
    }
    gcn_finalize<<<BATCH, 256, 0, stream>>>(x, nidx, W1, b1, W2, b2, W3, b3, Wl, bl,
                                            deg, cnt, me, msrc, out);
}